// GCL_CS_73478300500058
// MI455X (gfx1250) — compile-verified
//
#include <hip/hip_runtime.h>
#include <hip/hip_bf16.h>
#include <math.h>

// Problem constants (match reference)
#define NND   100000
#define DD    512
#define HH    512
#define NPAR  4096

typedef __attribute__((ext_vector_type(2))) float v2f;
typedef __attribute__((ext_vector_type(8))) float v8f;

// -------------------------------------------------------------------------
// zero fill (ints, for bin counters)
// -------------------------------------------------------------------------
__global__ void k_zero_i(int* __restrict__ p, int n) {
    int i = blockIdx.x * blockDim.x + threadIdx.x;
    if (i < n) p[i] = 0;
}

// -------------------------------------------------------------------------
// K1: h = x@W + b (pure GEMM, stores h only; q recomputed later)
// block = 256 thr (8 waves), tile 32 rows x 128 cols, 2 accs/wave.
// grid = (N/32, H/128)
// -------------------------------------------------------------------------
__global__ __launch_bounds__(256) void k_gemm_xw(
    const float* __restrict__ x, const float* __restrict__ W,
    const float* __restrict__ b, float* __restrict__ hout)
{
    __shared__ float As[32][68];    // stride 68: b64 frag reads hit all 64 banks once
    __shared__ float Bs[64][132];

    const int row0 = blockIdx.x * 32;
    const int col0 = blockIdx.y * 128;
    const int tid  = threadIdx.x;
    const int lane = tid & 31;          // wave32
    const int wave = tid >> 5;
    const int half = lane >> 4;
    const int lp   = lane & 15;

    v8f acc0 = {};
    v8f acc1 = {};

    for (int k0 = 0; k0 < DD; k0 += 64) {
        __syncthreads();
        #pragma unroll
        for (int i = 0; i < 2; ++i) {   // A tile: 32x64
            int idx = (tid + i * 256) * 4;
            int m = idx >> 6, kk = idx & 63;
            const float4 av = *(const float4*)(x + (size_t)(row0 + m) * DD + k0 + kk);
            As[m][kk+0]=av.x; As[m][kk+1]=av.y; As[m][kk+2]=av.z; As[m][kk+3]=av.w;
        }
        #pragma unroll
        for (int i = 0; i < 8; ++i) {   // B tile: 64x128
            int idx = (tid + i * 256) * 4;
            int kk = idx >> 7, c = idx & 127;
            const float4 bv = *(const float4*)(W + (size_t)(k0 + kk) * HH + col0 + c);
            Bs[kk][c+0]=bv.x; Bs[kk][c+1]=bv.y; Bs[kk][c+2]=bv.z; Bs[kk][c+3]=bv.w;
        }
        __syncthreads();

        #pragma unroll
        for (int kk = 0; kk < 64; kk += 4) {
            // A 16x4 layout: lanes 0-15 -> K={kk,kk+1}, lanes 16-31 -> K={kk+2,kk+3}
            const int ks = kk + 2 * half;
            const v2f a0 = *(const v2f*)&As[lp     ][ks];   // rows 0..15
            const v2f a1 = *(const v2f*)&As[lp + 16][ks];   // rows 16..31
            const int n = wave * 16 + lp;
            v2f bb;
            bb.x = Bs[ks    ][n];
            bb.y = Bs[ks + 1][n];
            acc0 = __builtin_amdgcn_wmma_f32_16x16x4_f32(
                false, a0, false, bb, (short)0, acc0, false, false);
            acc1 = __builtin_amdgcn_wmma_f32_16x16x4_f32(
                false, a1, false, bb, (short)0, acc1, false, false);
        }
    }

    const int col  = col0 + wave * 16 + lp;
    const float bias = b[col];
    #pragma unroll
    for (int r = 0; r < 8; ++r) {
        const int m = r + (half ? 8 : 0);       // C layout: VGPR r -> row r / r+8
        hout[(size_t)(row0 + m     ) * HH + col] = acc0[r] + bias;
        hout[(size_t)(row0 + 16 + m) * HH + col] = acc1[r] + bias;
    }
}

// -------------------------------------------------------------------------
// Binning: count -> scan -> scatter (replaces 102M float atomics)
// -------------------------------------------------------------------------
__global__ void k_count(const int* __restrict__ par, int* __restrict__ cnt) {
    int i = blockIdx.x * blockDim.x + threadIdx.x;
    if (i < NND) atomicAdd(&cnt[par[i]], 1);
}

__global__ void k_scan(const int* __restrict__ cnt, int* __restrict__ start) {
    if (threadIdx.x == 0 && blockIdx.x == 0) {
        int s = 0;
        for (int p = 0; p < NPAR; ++p) { start[p] = s; s += cnt[p]; }
        start[NPAR] = s;
    }
}

__global__ void k_scatter(const int* __restrict__ par, const int* __restrict__ start,
                          int* __restrict__ cursor, int* __restrict__ list) {
    int i = blockIdx.x * blockDim.x + threadIdx.x;
    if (i < NND) {
        const int p = par[i];
        const int slot = atomicAdd(&cursor[p], 1);
        list[start[p] + slot] = i;
    }
}

// -------------------------------------------------------------------------
// K_seg: per-parent gather-sum. k_p[p] = sum pw*h ; k_n[p] = sum pw*sigmoid(h)
// grid = (NPAR, H/128), block = 128 (one col per thread). Coalesced, no atomics.
// -------------------------------------------------------------------------
__global__ __launch_bounds__(128) void k_seg(
    const float* __restrict__ h, const float* __restrict__ pw,
    const int* __restrict__ list, const int* __restrict__ start,
    float* __restrict__ kp, float* __restrict__ kn)
{
    const int p   = blockIdx.x;
    const int col = blockIdx.y * 128 + threadIdx.x;
    const int s = start[p], e = start[p + 1];
    float ah = 0.0f, aq = 0.0f;
    for (int it = s; it < e; ++it) {
        const int   n  = list[it];
        const float w  = pw[n];
        const float hv = h[(size_t)n * HH + col];
        const float qv = 1.0f / (1.0f + __expf(-hv));
        ah += w * hv;
        aq += w * qv;
    }
    kp[(size_t)p * HH + col] = ah;
    kn[(size_t)p * HH + col] = aq;
}

// -------------------------------------------------------------------------
// K2: row-wise L2 normalize k_p (clamped at EPS). grid = NPAR blocks
// -------------------------------------------------------------------------
__global__ __launch_bounds__(256) void k_norm(float* __restrict__ kp)
{
    __shared__ float red[256];
    float* p = kp + (size_t)blockIdx.x * HH;
    float s = 0.0f;
    for (int i = threadIdx.x; i < HH; i += 256) { float v = p[i]; s += v * v; }
    red[threadIdx.x] = s;
    __syncthreads();
    for (int off = 128; off > 0; off >>= 1) {
        if (threadIdx.x < off) red[threadIdx.x] += red[threadIdx.x + off];
        __syncthreads();
    }
    const float scale = 1.0f / fmaxf(sqrtf(red[0]), 1e-12f);
    for (int i = threadIdx.x; i < HH; i += 256) p[i] *= scale;
}

// -------------------------------------------------------------------------
// K3: att = exp(2 * k_p @ k_p^T). tile 32x128. grid = (NPAR/32, NPAR/128)
// -------------------------------------------------------------------------
__global__ __launch_bounds__(256) void k_att(
    const float* __restrict__ kp, float* __restrict__ att)
{
    __shared__ float As[32][68];
    __shared__ float Bs[64][132];

    const int row0 = blockIdx.x * 32;
    const int col0 = blockIdx.y * 128;
    const int tid  = threadIdx.x;
    const int lane = tid & 31, wave = tid >> 5;
    const int half = lane >> 4, lp = lane & 15;

    v8f acc0 = {};
    v8f acc1 = {};

    for (int k0 = 0; k0 < DD; k0 += 64) {
        __syncthreads();
        #pragma unroll
        for (int i = 0; i < 2; ++i) {   // A: rows of k_p, 32x64
            int idx = (tid + i * 256) * 4;
            int m = idx >> 6, kk = idx & 63;
            const float4 av = *(const float4*)(kp + (size_t)(row0 + m) * DD + k0 + kk);
            As[m][kk+0]=av.x; As[m][kk+1]=av.y; As[m][kk+2]=av.z; As[m][kk+3]=av.w;
        }
        #pragma unroll
        for (int i = 0; i < 8; ++i) {   // B: k_p rows -> transpose on LDS store
            int idx = (tid + i * 256) * 4;
            int j = idx >> 6, kk = idx & 63;
            const float4 bv = *(const float4*)(kp + (size_t)(col0 + j) * DD + k0 + kk);
            Bs[kk+0][j]=bv.x; Bs[kk+1][j]=bv.y; Bs[kk+2][j]=bv.z; Bs[kk+3][j]=bv.w;
        }
        __syncthreads();

        #pragma unroll
        for (int kk = 0; kk < 64; kk += 4) {
            const int ks = kk + 2 * half;
            const v2f a0 = *(const v2f*)&As[lp     ][ks];
            const v2f a1 = *(const v2f*)&As[lp + 16][ks];
            const int n = wave * 16 + lp;
            v2f bb;
            bb.x = Bs[ks    ][n];
            bb.y = Bs[ks + 1][n];
            acc0 = __builtin_amdgcn_wmma_f32_16x16x4_f32(
                false, a0, false, bb, (short)0, acc0, false, false);
            acc1 = __builtin_amdgcn_wmma_f32_16x16x4_f32(
                false, a1, false, bb, (short)0, acc1, false, false);
        }
    }

    const int col = col0 + wave * 16 + lp;
    #pragma unroll
    for (int r = 0; r < 8; ++r) {
        const int m = r + (half ? 8 : 0);
        att[(size_t)(row0 + m     ) * NPAR + col] = expf(2.0f * acc0[r]);  // 1/T = 2
        att[(size_t)(row0 + 16 + m) * NPAR + col] = expf(2.0f * acc1[r]);
    }
}

// -------------------------------------------------------------------------
// K4: pos_msg = (att .* A_P) @ k_n ; neg_msg = att @ k_n
// tile 32x128, 4 accumulators/wave. grid = (NPAR/32, HH/128), K = NPAR
// -------------------------------------------------------------------------
__global__ __launch_bounds__(256) void k_msg(
    const float* __restrict__ att, const float* __restrict__ AP,
    const float* __restrict__ kn,
    float* __restrict__ posm, float* __restrict__ negm)
{
    __shared__ float Ts[32][68];   // att chunk
    __shared__ float Ps[32][68];   // A_P chunk
    __shared__ float Bs[64][132];  // k_n chunk

    const int row0 = blockIdx.x * 32;
    const int col0 = blockIdx.y * 128;
    const int tid  = threadIdx.x;
    const int lane = tid & 31, wave = tid >> 5;
    const int half = lane >> 4, lp = lane & 15;

    v8f accp0 = {}, accp1 = {};
    v8f accn0 = {}, accn1 = {};

    for (int k0 = 0; k0 < NPAR; k0 += 64) {
        __syncthreads();
        #pragma unroll
        for (int i = 0; i < 2; ++i) {   // att + A_P chunks: 32x64 each
            int idx = (tid + i * 256) * 4;
            int m = idx >> 6, kk = idx & 63;
            const size_t base = (size_t)(row0 + m) * NPAR + k0 + kk;
            const float4 tv = *(const float4*)(att + base);
            Ts[m][kk+0]=tv.x; Ts[m][kk+1]=tv.y; Ts[m][kk+2]=tv.z; Ts[m][kk+3]=tv.w;
            const float4 pv = *(const float4*)(AP + base);
            Ps[m][kk+0]=pv.x; Ps[m][kk+1]=pv.y; Ps[m][kk+2]=pv.z; Ps[m][kk+3]=pv.w;
        }
        #pragma unroll
        for (int i = 0; i < 8; ++i) {   // k_n chunk: 64x128
            int idx = (tid + i * 256) * 4;
            int kk = idx >> 7, c = idx & 127;
            const float4 bv = *(const float4*)(kn + (size_t)(k0 + kk) * HH + col0 + c);
            Bs[kk][c+0]=bv.x; Bs[kk][c+1]=bv.y; Bs[kk][c+2]=bv.z; Bs[kk][c+3]=bv.w;
        }
        __syncthreads();

        #pragma unroll
        for (int kk = 0; kk < 64; kk += 4) {
            const int ks = kk + 2 * half;
            const v2f an0 = *(const v2f*)&Ts[lp     ][ks];
            const v2f an1 = *(const v2f*)&Ts[lp + 16][ks];
            const v2f sc0 = *(const v2f*)&Ps[lp     ][ks];
            const v2f sc1 = *(const v2f*)&Ps[lp + 16][ks];
            const v2f ap0 = an0 * sc0;
            const v2f ap1 = an1 * sc1;
            const int n = wave * 16 + lp;
            v2f bb;
            bb.x = Bs[ks    ][n];
            bb.y = Bs[ks + 1][n];
            accn0 = __builtin_amdgcn_wmma_f32_16x16x4_f32(
                false, an0, false, bb, (short)0, accn0, false, false);
            accn1 = __builtin_amdgcn_wmma_f32_16x16x4_f32(
                false, an1, false, bb, (short)0, accn1, false, false);
            accp0 = __builtin_amdgcn_wmma_f32_16x16x4_f32(
                false, ap0, false, bb, (short)0, accp0, false, false);
            accp1 = __builtin_amdgcn_wmma_f32_16x16x4_f32(
                false, ap1, false, bb, (short)0, accp1, false, false);
        }
    }

    const int col = col0 + wave * 16 + lp;
    #pragma unroll
    for (int r = 0; r < 8; ++r) {
        const int m = r + (half ? 8 : 0);
        posm[(size_t)(row0 + m     ) * HH + col] = accp0[r];
        posm[(size_t)(row0 + 16 + m) * HH + col] = accp1[r];
        negm[(size_t)(row0 + m     ) * HH + col] = accn0[r];
        negm[(size_t)(row0 + 16 + m) * HH + col] = accn1[r];
    }
}

// -------------------------------------------------------------------------
// K5: per-node scores + log loss. One wave per node, 8 nodes/block.
// q recomputed from h (sigmoid), saving a 205MB q_n buffer.
// -------------------------------------------------------------------------
__global__ __launch_bounds__(256) void k_score(
    const float* __restrict__ h, const float* __restrict__ posm,
    const float* __restrict__ negm, const int* __restrict__ par,
    float* __restrict__ lossArr)
{
    const int node = blockIdx.x * 8 + (threadIdx.x >> 5);
    const int lane = threadIdx.x & 31;
    const int p = par[node];
    const float* hr = h    + (size_t)node * HH;
    const float* pm = posm + (size_t)p * HH;
    const float* nm = negm + (size_t)p * HH;
    float sp = 0.0f, sn = 0.0f;
    for (int i = lane; i < HH; i += 32) {
        const float qv = 1.0f / (1.0f + __expf(-hr[i]));
        sp += qv * pm[i];
        sn += qv * nm[i];
    }
    #pragma unroll
    for (int off = 16; off > 0; off >>= 1) {
        sp += __shfl_xor(sp, off);
        sn += __shfl_xor(sn, off);
    }
    if (lane == 0) lossArr[node] = logf(sn) - logf(sp);
}

// -------------------------------------------------------------------------
// K6: deterministic final mean reduce. Single block.
// -------------------------------------------------------------------------
__global__ __launch_bounds__(256) void k_reduce(
    const float* __restrict__ lossArr, float* __restrict__ out)
{
    __shared__ float red[256];
    float s = 0.0f;
    for (int i = threadIdx.x; i < NND; i += 256) s += lossArr[i];
    red[threadIdx.x] = s;
    __syncthreads();
    for (int off = 128; off > 0; off >>= 1) {
        if (threadIdx.x < off) red[threadIdx.x] += red[threadIdx.x + off];
        __syncthreads();
    }
    if (threadIdx.x == 0) out[0] = red[0] / (float)NND;
}

// -------------------------------------------------------------------------
extern "C" void kernel_launch(void* const* d_in, const int* in_sizes, int n_in,
                              void* d_out, int out_size, void* d_ws, size_t ws_size,
                              hipStream_t stream)
{
    const float* x   = (const float*)d_in[0];   // (N, D)
    const int*   ntp = (const int*)  d_in[1];   // (N,)
    const float* pwt = (const float*)d_in[2];   // (N,)
    const float* AP  = (const float*)d_in[3];   // (P, P)
    const float* W   = (const float*)d_in[4];   // (D, H)
    const float* b   = (const float*)d_in[5];   // (H,)

    float* ws = (float*)d_ws;
    size_t off = 0;
    float* hbuf    = ws + off; off += (size_t)NND * HH;     // 51.2M f
    float* kp      = ws + off; off += (size_t)NPAR * HH;    //  2.1M f
    float* kn      = ws + off; off += (size_t)NPAR * HH;
    float* att     = ws + off; off += (size_t)NPAR * NPAR;  // 16.8M f
    float* posm    = ws + off; off += (size_t)NPAR * HH;
    float* negm    = ws + off; off += (size_t)NPAR * HH;
    float* lossArr = ws + off; off += (size_t)NND;
    int* ibase  = (int*)(ws + off);
    int* cnt    = ibase;                 // NPAR
    int* cursor = ibase + NPAR;          // NPAR
    int* start  = ibase + 2 * NPAR;      // NPAR+1
    int* list   = ibase + 3 * NPAR + 1;  // NND

    // main GEMM: h = x@W + b
    k_gemm_xw<<<dim3(NND / 32, HH / 128), dim3(256), 0, stream>>>(x, W, b, hbuf);

    // bin nodes by parent (cnt+cursor zeroed together: contiguous 2*NPAR)
    k_zero_i<<<dim3((2 * NPAR + 255) / 256), dim3(256), 0, stream>>>(cnt, 2 * NPAR);
    k_count<<<dim3((NND + 255) / 256), dim3(256), 0, stream>>>(ntp, cnt);
    k_scan<<<dim3(1), dim3(64), 0, stream>>>(cnt, start);
    k_scatter<<<dim3((NND + 255) / 256), dim3(256), 0, stream>>>(ntp, start, cursor, list);

    // segment sums (gather, no float atomics)
    k_seg<<<dim3(NPAR, HH / 128), dim3(128), 0, stream>>>(hbuf, pwt, list, start, kp, kn);

    k_norm<<<dim3(NPAR), dim3(256), 0, stream>>>(kp);

    k_att<<<dim3(NPAR / 32, NPAR / 128), dim3(256), 0, stream>>>(kp, att);

    k_msg<<<dim3(NPAR / 32, HH / 128), dim3(256), 0, stream>>>(att, AP, kn, posm, negm);

    k_score<<<dim3(NND / 8), dim3(256), 0, stream>>>(hbuf, posm, negm, ntp, lossArr);

    k_reduce<<<dim3(1), dim3(256), 0, stream>>>(lossArr, (float*)d_out);
}